// LocalConvolution_56796647523036
// MI455X (gfx1250) — compile-verified
//
#include <hip/hip_runtime.h>
#include <stdint.h>

// Problem constants (from reference):
// x:      (2, 64, 16, 64, 64)  f32
// weight: (2, 1, 16, 27, 16, 64, 64) f32
// out:    (2, 64, 16, 64, 64)  f32
// out[b, q*16+c, d, h, w] = sum_k x[b, q*16+c, d+kd-1, h+kh-1, w+kw-1] * weight[b,0,c,k,d,h,w]
//
// Roofline: 0.45 GFLOP vs ~293 MB mandatory traffic -> pure HBM-bandwidth bound
// (~12.6 us floor at 23.3 TB/s). Weights (226.5 MB) are streamed once, non-temporal;
// x (33.5 MB) is staged via CDNA5 async DMA into LDS and kept L2-resident.

#define ROWLEN 72              // 64 interior + halos, padded so interior (col 4) is 16B aligned
#define NLINES (4 * 3 * 6)     // q(4) x dplane(3) x rows(6)
#define LDS_FLOATS (NLINES * ROWLEN)

__global__ __launch_bounds__(256) void localconv_kernel(
    const float* __restrict__ x,
    const float* __restrict__ weight,
    float* __restrict__ out)
{
  __shared__ __align__(16) float xs[LDS_FLOATS];

  const int tid = threadIdx.x;
  const int bix = blockIdx.x;           // 8192 blocks
  const int ht  = bix & 15;             // h tile (4 rows each)
  const int d   = (bix >> 4) & 15;
  const int c   = (bix >> 8) & 15;      // c within wc=16
  const int b   = bix >> 12;
  const int h0  = ht << 2;

  // ---- Phase 1: zero the LDS tile (covers padding, halos, OOB lines) ----
  for (int i = tid; i < LDS_FLOATS; i += 256) xs[i] = 0.0f;
  __syncthreads();  // zeros committed before any async LDS writes are issued

  // ---- Phase 2: async-stage x halo tile: [q=4][dp=3][row=6][64 cols] ----
  // 1152 16-byte chunks; each lane issues ~4-5 async b128 ops (CDNA5 ASYNCcnt path).
  for (int e = tid; e < NLINES * 16; e += 256) {
    const int line = e >> 4;
    const int col4 = (e & 15) << 2;          // 0,4,...,60 (floats)
    const int q  = line / 18;
    const int rm = line - q * 18;
    const int dp = rm / 6;
    const int r  = rm - dp * 6;
    const int dd = d - 1 + dp;               // input d plane
    const int hh = h0 - 1 + r;               // input h row
    if ((unsigned)dd < 16u && (unsigned)hh < 64u) {
      // byte offset from x base (fits easily in 32 bits: x is 33.5 MB)
      const unsigned g_off =
          ((unsigned)((((b * 64 + q * 16 + c) * 16 + dd) << 12) + (hh << 6) + col4)) * 4u;
      // LDS byte address = low 32 bits of flat pointer into the shared array
      const unsigned lds_b = (unsigned)(uintptr_t)(&xs[line * ROWLEN + 4 + col4]);
      // GVS-mode async DMA: each lane copies 16 bytes from global base saddr plus offset vaddr to the LDS address in vdst
      asm volatile("global_load_async_to_lds_b128 %0, %1, %2"
                   :
                   : "v"(lds_b), "v"(g_off), "s"(x)
                   : "memory");
    }
  }
#if __has_builtin(__builtin_amdgcn_s_wait_asynccnt)
  __builtin_amdgcn_s_wait_asynccnt(0);       // this wave's async copies complete
#else
  asm volatile("s_wait_asynccnt 0" ::: "memory");
#endif
  __syncthreads();                           // all waves' copies visible

  // ---- Phase 3: compute. Thread = (hs, w); produces Q=4 outputs. ----
  const int hs = tid >> 6;        // 0..3
  const int w  = tid & 63;
  const int hh = h0 + hs;

  // Weight stream: 27 unique, coalesced dwords per thread (dominant 226.5 MB traffic,
  // read exactly once since all 4 q reuse the same weights from registers).
  // Non-temporal so the once-read weight stream doesn't evict L2-resident x.
  const float* wp = weight
      + (size_t)(b * 16 + c) * 27 * 65536   // (b*wc + c) * kvol * D*H*W
      + ((size_t)d << 12) + (hh << 6) + w;
  __builtin_prefetch(wp, 0, 0);             // gfx1250 global_prefetch_b8 (NT-ish locality)

  float wk[27];
#pragma unroll
  for (int k = 0; k < 27; ++k)
    wk[k] = __builtin_nontemporal_load(wp + ((size_t)k << 16));   // k-stride = 64K floats

  float acc[4] = {0.f, 0.f, 0.f, 0.f};
#pragma unroll
  for (int q = 0; q < 4; ++q) {
#pragma unroll
    for (int dp = 0; dp < 3; ++dp) {
#pragma unroll
      for (int rr = 0; rr < 3; ++rr) {
        // output row hh reads input rows hh-1+rr -> LDS row hs+rr (row 0 == h0-1)
        const float* rowp = &xs[((q * 3 + dp) * 6 + (hs + rr)) * ROWLEN + 3 + w];
        const int kb = (dp * 3 + rr) * 3;
        acc[q] += rowp[0] * wk[kb + 0];
        acc[q] += rowp[1] * wk[kb + 1];
        acc[q] += rowp[2] * wk[kb + 2];
      }
    }
  }

  // ---- Store 4 outputs (channels q*16+c), coalesced, non-temporal (write-once) ----
  float* op = out + ((size_t)(b * 64 + c) * 16 + d) * 4096 + (hh << 6) + w;
#pragma unroll
  for (int q = 0; q < 4; ++q)
    __builtin_nontemporal_store(acc[q], op + (size_t)q * 1048576);  // 16*65536 floats per q
}

extern "C" void kernel_launch(void* const* d_in, const int* in_sizes, int n_in,
                              void* d_out, int out_size, void* d_ws, size_t ws_size,
                              hipStream_t stream) {
  (void)in_sizes; (void)n_in; (void)out_size; (void)d_ws; (void)ws_size;
  const float* x  = (const float*)d_in[0];
  const float* wt = (const float*)d_in[1];
  float* out = (float*)d_out;
  // grid: b(2) * c(16) * d(16) * htile(16) = 8192 blocks of 256 threads
  localconv_kernel<<<dim3(2 * 16 * 16 * 16), dim3(256), 0, stream>>>(x, wt, out);
}